// MutualNearestNeighborMatcher_21423296873107
// MI455X (gfx1250) — compile-verified
//
#include <hip/hip_runtime.h>
#include <stdint.h>

#define BB    8
#define NN    4096
#define MM    4096
#define NP1   4097
#define MP1   4097
#define KOUT  2048
#define THRESH 0.1f

typedef unsigned long long u64;
typedef unsigned int       u32;

// 16-byte payload with 4-byte alignment: row stride is 4097 floats, so rows are
// only dword-aligned. gfx12 unaligned-access mode still gives b128 loads.
struct __attribute__((packed, aligned(4))) F4U { float x, y, z, w; };

// ---- monotonic float <-> u32 ordering (handles the -1.0 invalid score) ----
__device__ __forceinline__ u32 ordf(float f) {
  u32 u = __float_as_uint(f);
  return (u & 0x80000000u) ? ~u : (u | 0x80000000u);
}
__device__ __forceinline__ float unordf(u32 u) {
  u32 b = (u & 0x80000000u) ? (u & 0x7FFFFFFFu) : ~u;
  return __uint_as_float(b);
}
// key = (ordered value << 32) | (~index)  -> u64 max == (max value, min index)
__device__ __forceinline__ u64 packkey(u32 ordval, u32 idx) {
  return ((u64)ordval << 32) | (u64)(0xFFFFFFFFu - idx);
}
__device__ __forceinline__ u32 key_idx(u64 k) { return 0xFFFFFFFFu - (u32)(k & 0xFFFFFFFFull); }
__device__ __forceinline__ float key_val(u64 k) { return unordf((u32)(k >> 32)); }
__device__ __forceinline__ u64 umax64(u64 a, u64 b) { return a > b ? a : b; }

__device__ __forceinline__ u64 shflxor64(u64 v, int m) {
  int lo = (int)(u32)v;
  int hi = (int)(u32)(v >> 32);
  lo = __shfl_xor(lo, m, 32);
  hi = __shfl_xor(hi, m, 32);
  return ((u64)(u32)hi << 32) | (u64)(u32)lo;
}

// ---- CDNA5 async global->LDS copy (ASYNCcnt-tracked, ISA ch. 10 async ops) ----
// For a generic pointer into the shared aperture, the LDS byte offset is the
// low 32 address bits.
__device__ __forceinline__ u32 lds_byte_off(const void* p) {
  return (u32)(uintptr_t)p;
}
__device__ __forceinline__ void async_copy_b128(u32 lds_off, const void* gsrc) {
  asm volatile("global_load_async_to_lds_b128 %0, %1, off"
               :: "v"(lds_off), "v"((u64)(uintptr_t)gsrc)
               : "memory");
}
__device__ __forceinline__ void wait_asynccnt0() {
  asm volatile("s_wait_asynccnt 0" ::: "memory");
}

// ---------------------------------------------------------------------------
// Kernel 0: init per-(batch,column) argmax table. key(0.0f, i=0) is the
// identity for P in [0,1) and reproduces argmax==0 for an all-zero column.
// ---------------------------------------------------------------------------
__global__ void mnn_init_cols(u64* __restrict__ colTab) {
  int g = blockIdx.x * blockDim.x + threadIdx.x;
  if (g < BB * MM) colTab[g] = packkey(ordf(0.0f), 0u);
}

// ---------------------------------------------------------------------------
// Kernel 1: single fused pass over P_core = P[:, :4096, :4096].
// Block = (batch, 64 rows). Thread t owns columns {4t + 1024k + e}.
// Row argmax: packed-u64 wave32 reduction + LDS combine (one barrier per row,
// double-buffered LDS slots). Column argmax: register partials merged with
// global_atomic_max_u64 (order-independent => deterministic).
// P is read exactly once: 537 MB -> ~23 us at 23.3 TB/s (HBM-bound floor).
// ---------------------------------------------------------------------------
__global__ __launch_bounds__(256) void mnn_scan(const float* __restrict__ P,
                                                u64* __restrict__ rowKey,
                                                u64* __restrict__ colTab) {
  const int t    = threadIdx.x;
  const int b    = blockIdx.x >> 6;          // 64 blocks per batch
  const int row0 = (blockIdx.x & 63) << 6;   // 64 rows per block

  __shared__ u64 wred[16];                   // 8 waves x 2 (double buffer)

  u64 colreg[16];
#pragma unroll
  for (int q = 0; q < 16; ++q) colreg[q] = 0ull;   // < any real key

  const size_t slab = (size_t)b * NP1 * MP1;

  for (int i = row0; i < row0 + 64; ++i) {
    const float* __restrict__ row = P + slab + (size_t)i * MP1;

    // CDNA5 prefetch path (global_prefetch_b8) for the next row's segment
    if (i + 1 < row0 + 64) __builtin_prefetch(row + MP1 + 4 * t, 0, 0);

    u64 rp = 0ull;
#pragma unroll
    for (int k = 0; k < 4; ++k) {
      const int c0 = 4 * t + 1024 * k;
      F4U v = *(const F4U*)(row + c0);
      float vals[4] = {v.x, v.y, v.z, v.w};
#pragma unroll
      for (int e = 0; e < 4; ++e) {
        u32 ov = ordf(vals[e]);
        rp = umax64(rp, packkey(ov, (u32)(c0 + e)));
        colreg[k * 4 + e] = umax64(colreg[k * 4 + e], packkey(ov, (u32)i));
      }
    }

    // wave32 argmax reduction
#pragma unroll
    for (int off = 16; off > 0; off >>= 1) rp = umax64(rp, shflxor64(rp, off));

    const int wid  = t >> 5;
    const int lane = t & 31;
    const int buf  = (i & 1) * 8;
    if (lane == 0) wred[buf + wid] = rp;
    __syncthreads();
    if (t == 0) {
      u64 r = wred[buf];
      for (int w = 1; w < 8; ++w) r = umax64(r, wred[buf + w]);
      rowKey[(size_t)b * NN + i] = r;
    }
    // no second barrier: next iteration uses the other wred buffer
  }

  // merge column partials into the global table
#pragma unroll
  for (int k = 0; k < 4; ++k) {
#pragma unroll
    for (int e = 0; e < 4; ++e) {
      int c = 4 * t + 1024 * k + e;
      atomicMax(&colTab[(size_t)b * MM + c], colreg[k * 4 + e]);
    }
  }
}

// ---------------------------------------------------------------------------
// Kernel 2: per batch — mutuality check, threshold, stable top-2048 via a
// full bitonic sort of 4096 packed keys in LDS, then gathers.
// The per-batch rowKey table (32 KB, 16B-aligned) is needed twice (key build
// + post-sort j-gather), so it is staged into LDS once with the CDNA5 async
// copy engine (global_load_async_to_lds_b128 + s_wait_asynccnt).
// ---------------------------------------------------------------------------
__global__ __launch_bounds__(1024) void mnn_match_sort(
    const u64* __restrict__ rowKey, const u64* __restrict__ colTab,
    const float* __restrict__ kp1, const float* __restrict__ kp2,
    float* __restrict__ out) {
  __shared__ u64 sk[NN];   // sort keys            (32 KB)
  __shared__ u64 rk[NN];   // staged rowKey table  (32 KB)
  const int t = threadIdx.x;
  const int b = blockIdx.x;

  // async-stage rowKey[b] into LDS: 2048 x b128 chunks, 2 per thread.
  const u64* gsrc = rowKey + (size_t)b * NN;
#pragma unroll
  for (int q = 0; q < 2; ++q) {
    int ch = t + q * 1024;                       // 16-byte chunk id
    async_copy_b128(lds_byte_off(&rk[2 * ch]), gsrc + 2 * ch);
  }
  wait_asynccnt0();
  __syncthreads();

#pragma unroll
  for (int q = 0; q < 4; ++q) {
    int i   = t + q * 1024;
    u64 rkk = rk[i];
    float p = key_val(rkk);
    u32 j   = key_idx(rkk);
    u32 ist = key_idx(colTab[(size_t)b * MM + j]);
    bool valid  = (ist == (u32)i) && (p >= THRESH);
    float score = valid ? p : -1.0f;
    sk[i] = packkey(ordf(score), (u32)i);
  }
  __syncthreads();

  // bitonic sort, descending
  for (int k = 2; k <= NN; k <<= 1) {
    for (int j = k >> 1; j > 0; j >>= 1) {
#pragma unroll
      for (int q = 0; q < 4; ++q) {
        int idx = t + q * 1024;
        int p   = idx ^ j;
        if (p > idx) {
          u64 a = sk[idx], c = sk[p];
          bool desc = ((idx & k) == 0);
          if (desc ? (a < c) : (a > c)) { sk[idx] = c; sk[p] = a; }
        }
      }
      __syncthreads();
    }
  }

  // outputs, concatenated flat in reference return order
  float* okp1  = out;                          // (B, 2048, 2)
  float* okp2  = out + (size_t)BB * KOUT * 2;  // (B, 2048, 2)
  float* osc   = okp2 + (size_t)BB * KOUT * 2; // (B, 2048)
  float* omask = osc + (size_t)BB * KOUT;      // (B, 2048)

#pragma unroll
  for (int q = 0; q < 2; ++q) {
    int r = t + q * 1024;
    u64 key = sk[r];
    float score = key_val(key);
    u32 i = key_idx(key);
    if (i > NN - 1) i = NN - 1;
    u32 j = key_idx(rk[i]);                    // from LDS-staged table
    if (j > MM - 1) j = MM - 1;
    size_t o = (size_t)b * KOUT + r;
    okp1[o * 2 + 0] = kp1[((size_t)b * NN + i) * 2 + 0];
    okp1[o * 2 + 1] = kp1[((size_t)b * NN + i) * 2 + 1];
    okp2[o * 2 + 0] = kp2[((size_t)b * MM + j) * 2 + 0];
    okp2[o * 2 + 1] = kp2[((size_t)b * MM + j) * 2 + 1];
    osc[o]   = score;
    omask[o] = (score > 0.0f) ? 1.0f : 0.0f;
  }
}

extern "C" void kernel_launch(void* const* d_in, const int* in_sizes, int n_in,
                              void* d_out, int out_size, void* d_ws, size_t ws_size,
                              hipStream_t stream) {
  const float* P   = (const float*)d_in[0];
  const float* kp1 = (const float*)d_in[1];
  const float* kp2 = (const float*)d_in[2];

  // workspace: colTab (8*4096 u64) | rowKey (8*4096 u64)  => 512 KB total
  // both 16B-aligned (hipMalloc base is 256B-aligned, offsets are multiples of 16)
  u64* colTab = (u64*)d_ws;
  u64* rowKey = colTab + (size_t)BB * MM;

  mnn_init_cols<<<(BB * MM) / 256, 256, 0, stream>>>(colTab);
  mnn_scan<<<BB * (NN / 64), 256, 0, stream>>>(P, rowKey, colTab);
  mnn_match_sort<<<BB, 1024, 0, stream>>>(rowKey, colTab, kp1, kp2, (float*)d_out);
}